// SinkhornDistance_43800076485251
// MI455X (gfx1250) — compile-verified
//
#include <hip/hip_runtime.h>
#include <math.h>

// ---------------------------------------------------------------------------
// Sinkhorn distance, B=8, N=1024, D=3, eps=0.1, <=100 iters, mean-EMD scalar.
//
// Phase 1 (WMMA): C[b,i,j] = max(|x_i|^2 + |y_j|^2 - 2*<x_i,y_j>, 0) into d_ws
//                 using V_WMMA_F32_16X16X4_F32. x/y staged in LDS pre-padded
//                 to K=4 so fragments are single unconditional ds_load_b64s.
// Phase 2: per-batch block (1024 thr) runs Sinkhorn iterations, u/v in LDS,
//          C streamed from L2-resident workspace; writes per-batch EMD.
// Phase 3: tiny kernel averages the 8 EMDs into d_out[0] (overwrite, no init).
// ---------------------------------------------------------------------------

#define NPTS      1024
#define NB        8
#define EPS_R     0.1f
#define INV_EPS   10.0f
#define MAX_ITERS 100
#define THRESH_C  0.1f
// log(1/1024 + 1e-8)
#define LOGMU    (-6.9314616f)

typedef float v2f __attribute__((ext_vector_type(2)));
typedef float v4f __attribute__((ext_vector_type(4)));
typedef float v8f __attribute__((ext_vector_type(8)));

// ------------------------------ Phase 1 ------------------------------------
// grid.x = 64 blocks (8 batches * 8 row-blocks), 256 threads (8 waves).
// Each wave owns one 16-row tile and sweeps all 64 column tiles with WMMA.
__global__ __launch_bounds__(256)
void sinkhorn_cost_wmma(const float* __restrict__ x,
                        const float* __restrict__ y,
                        float* __restrict__ C) {
    __shared__ float ys4[NPTS * 4];  // y padded to K=4 ([3]=0): 16 KB
    __shared__ float xs4[128 * 4];   // this block's 128 x rows, padded: 2 KB
    __shared__ float y2s[NPTS];      // |y_j|^2: 4 KB
    __shared__ float x2s[128];       // |x_i|^2 for block rows

    const int b    = blockIdx.x >> 3;   // batch
    const int rb   = blockIdx.x & 7;    // row-block (128 rows each)
    const int tid  = threadIdx.x;
    const int wave = tid >> 5;
    const int lane = tid & 31;

    const float* xb = x + (size_t)b * NPTS * 3;
    const float* yb = y + (size_t)b * NPTS * 3;
    float* Cb       = C + (size_t)b * NPTS * NPTS;

    // Stage y padded to 4 components (K=3 -> 4 with zero), plus |y|^2.
    for (int j = tid; j < NPTS; j += 256) {
        const float a0 = yb[j * 3 + 0], a1 = yb[j * 3 + 1], a2 = yb[j * 3 + 2];
        v4f p; p.x = a0; p.y = a1; p.z = a2; p.w = 0.0f;
        *(v4f*)&ys4[j * 4] = p;
        y2s[j] = a0 * a0 + a1 * a1 + a2 * a2;
    }
    // Stage this block's 128 x rows padded, plus |x|^2.
    if (tid < 128) {
        const int m = rb * 128 + tid;
        const float a0 = xb[m * 3 + 0], a1 = xb[m * 3 + 1], a2 = xb[m * 3 + 2];
        v4f p; p.x = a0; p.y = a1; p.z = a2; p.w = 0.0f;
        *(v4f*)&xs4[tid * 4] = p;
        x2s[tid] = a0 * a0 + a1 * a1 + a2 * a2;
    }
    __syncthreads();

    // Fragment layout (V_WMMA_F32_16X16X4_F32):
    //   lanes 0-15 carry K={0,1}, lanes 16-31 carry K={2,3}  (K=3 padded 0).
    const int koff = (lane >> 4) << 1;        // 0 for lo half, 2 for hi half
    const int la   = wave * 16 + (lane & 15); // local A row for this lane
    const v2f afrag = *(const v2f*)&xs4[la * 4 + koff];   // one ds_load_b64

    // D layout: VGPR r, lanes 0-15 -> M=r, lanes 16-31 -> M=r+8.
    const int rloc_base = wave * 16 + ((lane >> 4) & 1) * 8;
    const int ncol      = lane & 15;

    for (int nj = 0; nj < 64; ++nj) {
        const int bn = nj * 16 + (lane & 15);
        // B fragment: B[k][n] = y[n][k]; single unconditional ds_load_b64.
        const v2f bfrag = *(const v2f*)&ys4[bn * 4 + koff];

        v8f acc = {};
        acc = __builtin_amdgcn_wmma_f32_16x16x4_f32(
                  /*neg_a=*/false, afrag,
                  /*neg_b=*/false, bfrag,
                  /*c_mod=*/(short)0, acc,
                  /*reuse_a=*/false, /*reuse_b=*/false);

        const int   N   = nj * 16 + ncol;
        const float y2v = y2s[N];
        #pragma unroll
        for (int r = 0; r < 8; ++r) {
            const int Mloc = rloc_base + r;
            float c = x2s[Mloc] + y2v - 2.0f * acc[r];
            c = fmaxf(c, 0.0f);
            const int M = rb * 128 + Mloc;
            Cb[(size_t)M * NPTS + N] = c;
        }
    }
}

// ------------------------------ Phase 2 ------------------------------------
// One block (1024 threads = 32 waves) per batch. C streamed from workspace
// (L2 resident), u/v in LDS.
__global__ __launch_bounds__(1024)
void sinkhorn_iterate(const float* __restrict__ C,
                      float* __restrict__ emd_out) {
    __shared__ float u_s[NPTS];
    __shared__ float v_s[NPTS];
    __shared__ float red;
    __shared__ int   done_s;

    const int b    = blockIdx.x;
    const float* Cb = C + (size_t)b * NPTS * NPTS;
    const int tid  = threadIdx.x;
    const int wave = tid >> 5;
    const int lane = tid & 31;

    u_s[tid] = 0.0f;
    v_s[tid] = 0.0f;
    if (tid == 0) done_s = 0;
    __syncthreads();

    for (int it = 0; it < MAX_ITERS; ++it) {
        if (tid == 0) red = 0.0f;
        __syncthreads();

        // ---- u-update: wave w handles rows [w*32, w*32+32); lanes stride j.
        float wave_err = 0.0f;
        for (int rr = 0; rr < 32; ++rr) {
            const int i   = wave * 32 + rr;
            const float ui = u_s[i];
            float m = -3.0e38f, s = 0.0f;           // online logsumexp
            for (int j = lane; j < NPTS; j += 32) {
                const float val =
                    (ui + v_s[j] - Cb[(size_t)i * NPTS + j]) * INV_EPS;
                if (val > m) { s = s * __expf(m - val) + 1.0f; m = val; }
                else         { s += __expf(val - m); }
            }
            #pragma unroll
            for (int off = 16; off > 0; off >>= 1) {
                const float mo = __shfl_xor(m, off);
                const float so = __shfl_xor(s, off);
                const float mn = fmaxf(m, mo);
                s = s * __expf(m - mn) + so * __expf(mo - mn);
                m = mn;
            }
            const float lse  = m + __logf(s);
            const float unew = EPS_R * (LOGMU - lse) + ui;
            if (lane == 0) {
                wave_err += fabsf(unew - ui);
                u_s[i] = unew;
            }
        }
        if (lane == 0) atomicAdd(&red, wave_err);
        __syncthreads();   // u_s complete before v-update; red complete

        // ---- v-update: lane-private column tid, loop over rows (coalesced).
        {
            const int j   = tid;
            const float vj = v_s[j];
            float m = -3.0e38f, s = 0.0f;
            for (int i = 0; i < NPTS; ++i) {
                const float val =
                    (u_s[i] + vj - Cb[(size_t)i * NPTS + j]) * INV_EPS;
                if (val > m) { s = s * __expf(m - val) + 1.0f; m = val; }
                else         { s += __expf(val - m); }
            }
            const float lse = m + __logf(s);
            v_s[j] = EPS_R * (LOGMU - lse) + vj;
        }
        __syncthreads();   // v_s complete; red stable
        if (tid == 0) done_s = (red * (1.0f / NPTS) < THRESH_C) ? 1 : 0;
        __syncthreads();
        if (done_s) break;
    }

    // ---- emd_b = sum_ij exp((u_i + v_j - C)/eps) * C
    if (tid == 0) red = 0.0f;
    __syncthreads();
    float acc = 0.0f;
    for (int rr = 0; rr < 32; ++rr) {
        const int i    = wave * 32 + rr;
        const float ui = u_s[i];
        for (int j = lane; j < NPTS; j += 32) {
            const float c = Cb[(size_t)i * NPTS + j];
            acc += __expf((ui + v_s[j] - c) * INV_EPS) * c;
        }
    }
    #pragma unroll
    for (int off = 16; off > 0; off >>= 1) acc += __shfl_xor(acc, off);
    if (lane == 0) atomicAdd(&red, acc);
    __syncthreads();
    if (tid == 0) emd_out[b] = red;
}

// ------------------------------ Phase 3 ------------------------------------
__global__ void sinkhorn_finalize(const float* __restrict__ emd,
                                  float* __restrict__ out) {
    if (threadIdx.x == 0) {
        float s = 0.0f;
        #pragma unroll
        for (int b = 0; b < NB; ++b) s += emd[b];
        out[0] = s * (1.0f / NB);   // overwrite: no zero-init dependence
    }
}

// ---------------------------------------------------------------------------
extern "C" void kernel_launch(void* const* d_in, const int* in_sizes, int n_in,
                              void* d_out, int out_size, void* d_ws,
                              size_t ws_size, hipStream_t stream) {
    const float* x = (const float*)d_in[0];
    const float* y = (const float*)d_in[1];

    float* C   = (float*)d_ws;                       // 8*1024*1024 f32 = 32 MB
    float* emd = C + (size_t)NB * NPTS * NPTS;       // + 8 f32

    sinkhorn_cost_wmma<<<64, 256, 0, stream>>>(x, y, C);
    sinkhorn_iterate <<<NB, 1024, 0, stream>>>(C, emd);
    sinkhorn_finalize<<<1, 32, 0, stream>>>(emd, (float*)d_out);
}